// Awakening_Prompt_15693810499608
// MI455X (gfx1250) — compile-verified
//
#include <hip/hip_runtime.h>
#include <math.h>

#define DIM 768
#define NUM_PROMPTS 10
#define LATENT 20
#define BATCH 16
#define NTOK 4096
#define ROWS_X 4107                 // NUM_PROMPTS + 1 + NTOK
#define TOT_X (BATCH * ROWS_X)      // 65712 rows
#define TOT_L (BATCH * NTOK)        // 65536 rows
#define TILES_X (TOT_X / 16)        // 4107
#define TILES_L (TOT_L / 16)        // 4096
#define KSTEPS_D (DIM / 4)          // 192
#define NT_UP (DIM / 16)            // 48
#define KSTEPS_U (LATENT / 4)       // 5
#define EPSV 1e-5f

typedef float v2f __attribute__((ext_vector_type(2)));
typedef float v8f __attribute__((ext_vector_type(8)));

__device__ __forceinline__ v8f wmma4(v2f a, v2f b, v8f c) {
  // D = A(16x4,f32) * B(4x16,f32) + C(16x16,f32)
  return __builtin_amdgcn_wmma_f32_16x16x4_f32(false, a, false, b, (short)0, c,
                                               false, false);
}

// Fast sigmoid via v_rcp_f32 (avoids the IEEE div_scale/div_fixup sequence).
__device__ __forceinline__ float fast_sigmoid(float x) {
  return __builtin_amdgcn_rcpf(1.f + __expf(-x));
}
__device__ __forceinline__ float quick_gelu(float v) {
  return v * fast_sigmoid(1.702f * v);
}

// ---------------------------------------------------------------------------
// Prep: reswizzle Wd (768x20 -> padded, WMMA-B layout) and Wu (20x768) into
// per-lane float2 entries so GEMM waves do one coalesced b64 per B operand.
// Bd_pre[((ks*2 + nt)*32 + lane)]  (12288 float2)
// Bu_pre[((nt*5 + ks)*32 + lane)]  ( 7680 float2)
// bdp[32] : bd padded with zeros
// ---------------------------------------------------------------------------
__global__ __launch_bounds__(256) void prep_kernel(
    const float* __restrict__ Wd, const float* __restrict__ bd,
    const float* __restrict__ Wu, float* __restrict__ Bd_pre,
    float* __restrict__ Bu_pre, float* __restrict__ bdp) {
  int gid = blockIdx.x * blockDim.x + threadIdx.x;
  const int ND = KSTEPS_D * 2 * 32;          // 12288
  const int NU = NT_UP * KSTEPS_U * 32;      // 7680
  if (gid < ND) {
    int lane = gid & 31;
    int nt = (gid >> 5) & 1;
    int ks = gid >> 6;
    int n = lane & 15, kh = lane >> 4;
    int col = nt * 16 + n;
    int k = ks * 4 + 2 * kh;
    float x0 = 0.f, x1 = 0.f;
    if (col < LATENT) {
      x0 = Wd[k * LATENT + col];
      x1 = Wd[(k + 1) * LATENT + col];
    }
    Bd_pre[gid * 2 + 0] = x0;
    Bd_pre[gid * 2 + 1] = x1;
  } else if (gid < ND + NU) {
    int g = gid - ND;
    int lane = g & 31;
    int ks = (g >> 5) % KSTEPS_U;
    int nt = (g >> 5) / KSTEPS_U;
    int n = lane & 15, kh = lane >> 4;
    int col = nt * 16 + n;
    int k = ks * 4 + 2 * kh;
    Bu_pre[g * 2 + 0] = Wu[k * DIM + col];
    Bu_pre[g * 2 + 1] = Wu[(k + 1) * DIM + col];
  } else if (gid < ND + NU + 32) {
    int c = gid - (ND + NU);
    bdp[c] = (c < LATENT) ? bd[c] : 0.f;
  }
}

// ---------------------------------------------------------------------------
// Down-projection + quickGELU: rows(x | local_tokens) @ Wd + bd
// One wave per 16-row tile; K=768 in 192 steps of V_WMMA_F32_16X16X4_F32.
// ---------------------------------------------------------------------------
__global__ __launch_bounds__(128) void downproj_kernel(
    const float* __restrict__ X, const float* __restrict__ Ltok,
    const float* __restrict__ Bd_pre, const float* __restrict__ bdp,
    float* __restrict__ xlat, float* __restrict__ llat) {
  int wave = threadIdx.x >> 5;
  int lane = threadIdx.x & 31;
  int tile = blockIdx.x * 4 + wave;
  if (tile >= TILES_X + TILES_L) return;  // wave-uniform
  const float* src;
  float* dst;
  if (tile < TILES_X) {
    src = X + (size_t)tile * 16 * DIM;
    dst = xlat + (size_t)tile * 16 * LATENT;
  } else {
    int t = tile - TILES_X;
    src = Ltok + (size_t)t * 16 * DIM;
    dst = llat + (size_t)t * 16 * LATENT;
  }
  int n = lane & 15;   // A: row m ; B/C/D: col n
  int kh = lane >> 4;  // lane-half selects K+2 (A/B), M+8 (C/D)

  const float* arow = src + (size_t)n * DIM + 2 * kh;
  float bias0 = bdp[n];
  float bias1 = bdp[16 + n];
  v8f c0, c1;
#pragma unroll
  for (int j = 0; j < 8; ++j) { c0[j] = bias0; c1[j] = bias1; }

  const v2f* bp = (const v2f*)Bd_pre;
#pragma unroll 4
  for (int ks = 0; ks < KSTEPS_D; ++ks) {
    v2f a = *(const v2f*)(arow + ks * 4);      // x[m][k+2kh], x[m][k+2kh+1]
    v2f b0 = bp[(ks * 2 + 0) * 32 + lane];
    v2f b1 = bp[(ks * 2 + 1) * 32 + lane];
    c0 = wmma4(a, b0, c0);
    c1 = wmma4(a, b1, c1);
  }
#pragma unroll
  for (int j = 0; j < 8; ++j) {
    int m = j + 8 * kh;
    dst[m * LATENT + n] = quick_gelu(c0[j]);
    if (n < 4) dst[m * LATENT + 16 + n] = quick_gelu(c1[j]);
  }
}

// ---------------------------------------------------------------------------
// Attention: one block per (batch, {global,local}, prompt). Online softmax
// over 4096 L2-resident token latents; LDS tree merge.
// ---------------------------------------------------------------------------
__global__ __launch_bounds__(256) void attn_kernel(
    const float* __restrict__ xlat, const float* __restrict__ llat,
    const float* __restrict__ Wgq, const float* __restrict__ bgq,
    const float* __restrict__ Wlq, const float* __restrict__ blq,
    float* __restrict__ ctxg, float* __restrict__ ctxl) {
  __shared__ float qv[LATENT];
  __shared__ float prow[LATENT];
  __shared__ float red_m[256], red_s[256];
  __shared__ float red_acc[256][21];  // pad 20->21: conflict-free

  int tid = threadIdx.x;
  int bx = blockIdx.x;
  int b = bx / 20;
  int r = bx % 20;
  int typ = r / NUM_PROMPTS;
  int p = r % NUM_PROMPTS;

  if (tid < LATENT)
    prow[tid] = xlat[((size_t)b * ROWS_X + p) * LATENT + tid];
  __syncthreads();
  if (tid < LATENT) {
    const float* Wq = typ ? Wlq : Wgq;
    const float* bq = typ ? blq : bgq;
    float v = bq[tid];
#pragma unroll
    for (int d = 0; d < LATENT; ++d) v += prow[d] * Wq[d * LATENT + tid];
    qv[tid] = v;
  }
  __syncthreads();

  const float* tok =
      typ ? (llat + (size_t)b * NTOK * LATENT)
          : (xlat + ((size_t)b * ROWS_X + NUM_PROMPTS + 1) * LATENT);
  const float scale = 0.22360679774997896f;  // LATENT^-0.5

  float m = -1e30f, s = 0.f;
  float acc[LATENT];
#pragma unroll
  for (int d = 0; d < LATENT; ++d) acc[d] = 0.f;

  for (int nn = tid; nn < NTOK; nn += 256) {
    const float4* t4 = (const float4*)(tok + (size_t)nn * LATENT);
    float tv[LATENT];
    float4 f;
    f = t4[0]; tv[0] = f.x; tv[1] = f.y; tv[2] = f.z; tv[3] = f.w;
    f = t4[1]; tv[4] = f.x; tv[5] = f.y; tv[6] = f.z; tv[7] = f.w;
    f = t4[2]; tv[8] = f.x; tv[9] = f.y; tv[10] = f.z; tv[11] = f.w;
    f = t4[3]; tv[12] = f.x; tv[13] = f.y; tv[14] = f.z; tv[15] = f.w;
    f = t4[4]; tv[16] = f.x; tv[17] = f.y; tv[18] = f.z; tv[19] = f.w;
    float dot = 0.f;
#pragma unroll
    for (int d = 0; d < LATENT; ++d) dot += qv[d] * tv[d];
    float w = dot * scale;
    float nm = fmaxf(m, w);
    float e1 = __expf(m - nm), e2 = __expf(w - nm);
    s = s * e1 + e2;
#pragma unroll
    for (int d = 0; d < LATENT; ++d) acc[d] = acc[d] * e1 + e2 * tv[d];
    m = nm;
  }

  red_m[tid] = m;
  red_s[tid] = s;
#pragma unroll
  for (int d = 0; d < LATENT; ++d) red_acc[tid][d] = acc[d];
  __syncthreads();
  for (int st = 128; st >= 1; st >>= 1) {
    if (tid < st) {
      float m1 = red_m[tid], m2 = red_m[tid + st];
      float nm = fmaxf(m1, m2);
      float e1 = __expf(m1 - nm), e2 = __expf(m2 - nm);
      red_s[tid] = red_s[tid] * e1 + red_s[tid + st] * e2;
#pragma unroll
      for (int d = 0; d < LATENT; ++d)
        red_acc[tid][d] = red_acc[tid][d] * e1 + red_acc[tid + st][d] * e2;
      red_m[tid] = nm;
    }
    __syncthreads();
  }
  if (tid < LATENT) {
    float* ctx = typ ? ctxl : ctxg;
    ctx[((size_t)b * NUM_PROMPTS + p) * LATENT + tid] =
        red_acc[0][tid] / red_s[0];
  }
}

// ---------------------------------------------------------------------------
// Fuse: layernorms, importance MLP (exact-erf GELU), gate, fuse contexts;
// writes enhanced prompts in place over x_latent rows 0..9 per batch.
// ---------------------------------------------------------------------------
__global__ __launch_bounds__(64) void fuse_kernel(
    float* __restrict__ xlat, const float* __restrict__ ctxg,
    const float* __restrict__ ctxl, const float* __restrict__ ln_c_g,
    const float* __restrict__ ln_c_b, const float* __restrict__ Wc1,
    const float* __restrict__ bc1, const float* __restrict__ Wc2,
    const float* __restrict__ bc2, const float* __restrict__ ln_g_g,
    const float* __restrict__ ln_g_b, const float* __restrict__ Wg,
    const float* __restrict__ bg) {
  __shared__ float cls[LATENT], clsc[LATENT], clsg[LATENT], h[64],
      imp[NUM_PROMPTS];
  __shared__ float stats[2];
  __shared__ float gws;
  int tid = threadIdx.x;
  int b = blockIdx.x;

  if (tid < LATENT)
    cls[tid] = xlat[((size_t)b * ROWS_X + NUM_PROMPTS) * LATENT + tid];
  __syncthreads();
  if (tid == 0) {
    float mu = 0.f;
    for (int d = 0; d < LATENT; ++d) mu += cls[d];
    mu /= (float)LATENT;
    float var = 0.f;
    for (int d = 0; d < LATENT; ++d) {
      float t = cls[d] - mu;
      var += t * t;
    }
    var /= (float)LATENT;
    stats[0] = mu;
    stats[1] = rsqrtf(var + EPSV);
  }
  __syncthreads();
  if (tid < LATENT) {
    float xn = (cls[tid] - stats[0]) * stats[1];
    clsc[tid] = xn * ln_c_g[tid] + ln_c_b[tid];
    clsg[tid] = xn * ln_g_g[tid] + ln_g_b[tid];
  }
  __syncthreads();
  {
    float a = bc1[tid];
    for (int d = 0; d < LATENT; ++d) a += clsc[d] * Wc1[d * 64 + tid];
    h[tid] = 0.5f * a * (1.f + erff(a * 0.70710678118654752f));  // exact GELU
  }
  __syncthreads();
  if (tid < NUM_PROMPTS) {
    float v = bc2[tid];
    for (int j = 0; j < 64; ++j) v += h[j] * Wc2[j * NUM_PROMPTS + tid];
    imp[tid] = fast_sigmoid(v);
  }
  if (tid == 32) {
    float v = bg[0];
    for (int d = 0; d < LATENT; ++d) v += clsg[d] * Wg[d];
    gws = fast_sigmoid(v);
  }
  __syncthreads();
  float gw = gws;
  for (int idx = tid; idx < NUM_PROMPTS * LATENT; idx += 64) {
    int p = idx / LATENT, d = idx % LATENT;
    float g = ctxg[((size_t)b * NUM_PROMPTS + p) * LATENT + d];
    float l = ctxl[((size_t)b * NUM_PROMPTS + p) * LATENT + d];
    float f = gw * g + (1.f - gw) * l;
    xlat[((size_t)b * ROWS_X + p) * LATENT + d] = f * imp[p];
  }
}

// ---------------------------------------------------------------------------
// Up-projection: combined(=x_latent) @ Wu + bu. A tile (16x20) kept in 5 v2f
// register pairs, reused across 48 N-tiles of WMMA.
// ---------------------------------------------------------------------------
__global__ __launch_bounds__(128) void upproj_kernel(
    const float* __restrict__ lat, const float* __restrict__ Bu_pre,
    const float* __restrict__ bu, float* __restrict__ out) {
  int wave = threadIdx.x >> 5;
  int lane = threadIdx.x & 31;
  int tile = blockIdx.x * 4 + wave;
  if (tile >= TILES_X) return;  // wave-uniform
  int n = lane & 15, kh = lane >> 4;

  const float* arow = lat + ((size_t)tile * 16 + n) * LATENT + 2 * kh;
  v2f a[KSTEPS_U];
#pragma unroll
  for (int s = 0; s < KSTEPS_U; ++s) a[s] = *(const v2f*)(arow + s * 4);

  const v2f* bp = (const v2f*)Bu_pre;
  float* orow = out + (size_t)tile * 16 * DIM;
  for (int nt = 0; nt < NT_UP; ++nt) {
    int col = nt * 16 + n;
    float bias = bu[col];
    v8f c;
#pragma unroll
    for (int j = 0; j < 8; ++j) c[j] = bias;
#pragma unroll
    for (int s = 0; s < KSTEPS_U; ++s)
      c = wmma4(a[s], bp[(nt * KSTEPS_U + s) * 32 + lane], c);
#pragma unroll
    for (int j = 0; j < 8; ++j) {
      int m = j + 8 * kh;
      orow[m * DIM + col] = c[j];
    }
  }
}

// ---------------------------------------------------------------------------
extern "C" void kernel_launch(void* const* d_in, const int* in_sizes, int n_in,
                              void* d_out, int out_size, void* d_ws,
                              size_t ws_size, hipStream_t stream) {
  const float* x = (const float*)d_in[0];
  const float* ltok = (const float*)d_in[1];
  const float* Wd = (const float*)d_in[2];
  const float* bd = (const float*)d_in[3];
  const float* Wu = (const float*)d_in[4];
  const float* bu = (const float*)d_in[5];
  const float* Wgq = (const float*)d_in[6];
  const float* bgq = (const float*)d_in[7];
  const float* Wlq = (const float*)d_in[8];
  const float* blq = (const float*)d_in[9];
  const float* ln_c_g = (const float*)d_in[10];
  const float* ln_c_b = (const float*)d_in[11];
  const float* Wc1 = (const float*)d_in[12];
  const float* bc1 = (const float*)d_in[13];
  const float* Wc2 = (const float*)d_in[14];
  const float* bc2 = (const float*)d_in[15];
  const float* ln_g_g = (const float*)d_in[16];
  const float* ln_g_b = (const float*)d_in[17];
  const float* Wg = (const float*)d_in[18];
  const float* bg = (const float*)d_in[19];
  float* out = (float*)d_out;

  float* ws = (float*)d_ws;
  float* xlat = ws;                                   // 65712*20
  float* llat = xlat + (size_t)TOT_X * LATENT;        // 65536*20
  float* ctxg = llat + (size_t)TOT_L * LATENT;        // 16*10*20
  float* ctxl = ctxg + BATCH * NUM_PROMPTS * LATENT;  // 16*10*20
  float* Bd_pre = ctxl + BATCH * NUM_PROMPTS * LATENT;     // 24576 f
  float* Bu_pre = Bd_pre + (size_t)KSTEPS_D * 2 * 32 * 2;  // 15360 f
  float* bdp = Bu_pre + (size_t)NT_UP * KSTEPS_U * 32 * 2; // 32 f

  int prep_total = KSTEPS_D * 2 * 32 + NT_UP * KSTEPS_U * 32 + 32;
  prep_kernel<<<(prep_total + 255) / 256, 256, 0, stream>>>(Wd, bd, Wu, Bd_pre,
                                                            Bu_pre, bdp);
  int tiles = TILES_X + TILES_L;
  downproj_kernel<<<(tiles + 3) / 4, 128, 0, stream>>>(x, ltok, Bd_pre, bdp,
                                                       xlat, llat);
  attn_kernel<<<BATCH * 2 * NUM_PROMPTS, 256, 0, stream>>>(
      xlat, llat, Wgq, bgq, Wlq, blq, ctxg, ctxl);
  fuse_kernel<<<BATCH, 64, 0, stream>>>(xlat, ctxg, ctxl, ln_c_g, ln_c_b, Wc1,
                                        bc1, Wc2, bc2, ln_g_g, ln_g_b, Wg, bg);
  upproj_kernel<<<(TILES_X + 3) / 4, 128, 0, stream>>>(xlat, Bu_pre, bu, out);
}